// Seq2Seq_68513318305936
// MI455X (gfx1250) — compile-verified
//
#include <hip/hip_runtime.h>
#include <math.h>

typedef __attribute__((ext_vector_type(2))) float v2f;
typedef __attribute__((ext_vector_type(8))) float v8f;
typedef __attribute__((ext_vector_type(4))) unsigned int u32x4;
typedef __attribute__((ext_vector_type(4))) int i32x4;
typedef __attribute__((ext_vector_type(8))) int i32x8;

#define HID   1024
#define GATES 4096
#define EMB   512
#define NB    16      // batch
#define S_ENC 128
#define T_DEC 65
#define VOCAB 32000

// ---------------------------------------------------------------- WMMA core
__device__ inline v8f wmma4(v2f a, v2f b, v8f c) {
  // V_WMMA_F32_16X16X4_F32 : D = A(16x4,f32) * B(4x16,f32) + C(16x16,f32)
  return __builtin_amdgcn_wmma_f32_16x16x4_f32(
      false, a, false, b, (short)0, c, false, false);
}

__device__ inline float sigmoidf_(float x) { return 1.0f / (1.0f + __expf(-x)); }

// ------------------------------------------------- TDM tile load (async DMA)
// Loads a [tileDim2?][tileRows][K] f32 tile (row stride K elems, plane stride
// dim1StrideElems) into LDS with 2 pad DWORDs every 256 DWORDs, so the LDS
// row stride is K + 2*(K/256) floats (keeps 8B alignment, staggers banks).
__device__ inline unsigned lds_off_of(const void* p) {
  return (unsigned)(uintptr_t)p;   // flat LDS addr: low 32 bits = LDS offset
}

__device__ inline void tdm_load(unsigned lds_off, const float* g, unsigned K,
                                unsigned tileRows, unsigned tensorRows,
                                unsigned tileDim2,
                                unsigned long long dim1StrideElems) {
  unsigned long long ga = (unsigned long long)(uintptr_t)g;
  u32x4 g0;
  g0.x = 1u;                                        // count=1 (valid user D#)
  g0.y = lds_off;                                   // lds_addr
  g0.z = (unsigned)(ga & 0xFFFFFFFFull);            // global_addr[31:0]
  g0.w = (unsigned)((ga >> 32) & 0x1FFFFFFull)      // global_addr[56:32]
       | (2u << 30);                                // type = 2 ("image")
  i32x8 g1;
  g1[0] = (int)((2u << 16)                          // data_size = 4B
              | (1u << 20)                          // pad_enable
              | (7u << 22)                          // pad_interval = 256 DW
              | (1u << 25));                        // pad_amount   = 2 DW
  g1[1] = (int)((K & 0xFFFFu) << 16);               // tensor_dim0[15:0]
  g1[2] = (int)((K >> 16) | ((tensorRows & 0xFFFFu) << 16));
  g1[3] = (int)((tensorRows >> 16) | ((K & 0xFFFFu) << 16));   // tile_dim0
  g1[4] = (int)((tileRows & 0xFFFFu) | ((tileDim2 & 0xFFFFu) << 16));
  g1[5] = (int)(K & 0xFFFFFFFFu);                   // tensor_dim0_stride
  g1[6] = (int)((unsigned)((dim1StrideElems & 0xFFFFull) << 16));
  g1[7] = (int)(dim1StrideElems >> 16);             // tensor_dim1_stride[47:16]
  i32x4 g2 = { (int)(tileDim2 ? tileDim2 : 1), 0, 0, 0 };   // tensor_dim2
  i32x4 g3 = { 0, 0, 0, 0 };
  i32x8 g4 = { 0, 0, 0, 0, 0, 0, 0, 0 };            // unused trailing group
  __builtin_amdgcn_tensor_load_to_lds(g0, g1, g2, g3, g4, 0);
}

// ------------------------------------------------------- grid-wide barrier
__device__ inline void grid_barrier(unsigned* bar, unsigned nb, unsigned gen) {
  __syncthreads();
  if (threadIdx.x == 0) {
    __threadfence();
    atomicAdd(bar, 1u);
    const unsigned target = gen * nb;                    // monotonic counter
    while (atomicAdd(bar, 0u) < target) { __builtin_amdgcn_s_sleep(1); }
    __threadfence();
  }
  __syncthreads();
}

// ------------------------------------------------------------- embeddings
__global__ void embed_kernel(const int* __restrict__ tokens,
                             const float* __restrict__ emb,
                             float* __restrict__ X, int S, int total4) {
  int idx = blockIdx.x * blockDim.x + threadIdx.x;
  if (idx >= total4) return;
  int e4  = idx & (EMB / 4 - 1);
  int row = idx >> 7;               // t*NB + b
  int b   = row & (NB - 1);
  int t   = row >> 4;
  int tok = tokens[b * S + t];
  ((float4*)(X + (size_t)row * EMB))[e4] =
      ((const float4*)(emb + (size_t)tok * EMB))[e4];
}

// -------------------------------------------------- shared-B GEMM kernel
// Y[M,N] = A[M,K] @ W[N,K]^T + bias.  One block per 64-col N tile: B tile is
// TDM-staged into LDS once and reused by all M tiles (W read once from HBM).
// outMode 0: Y row-major [M,N], bias = b0+b1   (input projections)
// outMode 1: out[b][t][v] layout, bias = b0    (vocab projection)
__global__ void __launch_bounds__(256, 1)
gemm_bT_kernel(const float* __restrict__ A, const float* __restrict__ W,
               const float* __restrict__ b0, const float* __restrict__ b1,
               float* __restrict__ Y, int M, int K, int N, int outMode) {
  extern __shared__ float bsh[];                  // [64][K + 2*(K/256)]
  const int lane  = threadIdx.x & 31;
  const int wave  = threadIdx.x >> 5;
  const int nBase = blockIdx.x << 6;
  const int PR    = K + ((K >> 8) << 1);          // padded LDS row stride

  if (threadIdx.x < 32) {                         // one wave issues the DMA
    tdm_load(lds_off_of(bsh), W + (size_t)nBase * K, (unsigned)K, 64u,
             (unsigned)N, 0u, 0ull);
    __builtin_amdgcn_s_wait_tensorcnt(0);
  }
  __syncthreads();

  const int m     = lane & 15;
  const int koff  = (lane >> 4) << 1;             // half-wave K pair
  const int rHalf = (lane >> 4) << 3;
  const float* Brow0 = bsh + (size_t)(m +  0) * PR;
  const float* Brow1 = bsh + (size_t)(m + 16) * PR;
  const float* Brow2 = bsh + (size_t)(m + 32) * PR;
  const float* Brow3 = bsh + (size_t)(m + 48) * PR;

  const int MT     = M >> 4;
  const int stride = (int)(gridDim.y << 3);
  for (int mT = (int)(blockIdx.y << 3) + wave; mT < MT; mT += stride) {
    const float* Arow = A + (size_t)(mT * 16 + m) * K + koff;
    v8f acc[4] = {v8f{}, v8f{}, v8f{}, v8f{}};
    for (int k = 0; k < K; k += 4) {
      __builtin_prefetch(Arow + k + 128, 0, 3);   // near-cache prefetch
      const int kp = k + koff + ((k >> 8) << 1);  // padded LDS col (even)
      // hoist independent fragments so ds_loads can be clustered
      v2f a   = *(const v2f*)(Arow + k);
      v2f bb0 = *(const v2f*)(Brow0 + kp);
      v2f bb1 = *(const v2f*)(Brow1 + kp);
      v2f bb2 = *(const v2f*)(Brow2 + kp);
      v2f bb3 = *(const v2f*)(Brow3 + kp);
      acc[0] = wmma4(a, bb0, acc[0]);
      acc[1] = wmma4(a, bb1, acc[1]);
      acc[2] = wmma4(a, bb2, acc[2]);
      acc[3] = wmma4(a, bb3, acc[3]);
    }
    if (outMode == 0) {
#pragma unroll
      for (int s = 0; s < 4; ++s) {
        const int col = nBase + s * 16 + m;
        const float bias = b0[col] + b1[col];
#pragma unroll
        for (int v = 0; v < 8; ++v)
          Y[(size_t)(mT * 16 + v + rHalf) * N + col] = acc[s][v] + bias;
      }
    } else {
#pragma unroll
      for (int s = 0; s < 4; ++s) {
        const int col = nBase + s * 16 + m;
        const float bias = b0[col];
#pragma unroll
        for (int v = 0; v < 8; ++v)
          Y[((size_t)(v + rHalf) * T_DEC + mT) * (size_t)N + col] =
              acc[s][v] + bias;
      }
    }
  }
}

// ------------------------------------------------------ persistent LSTM
// 256 threads = 8 waves (2/SIMD): wave = 2*gate + kHalf. Each wave computes a
// half-K (512) partial of its gate tile; activation sums the two partials.
__device__ inline void lstm_step(const float* __restrict__ Xp_t,
                                 const float* __restrict__ hOld,
                                 float* __restrict__ hNew,
                                 float* __restrict__ hsOut,
                                 const float* __restrict__ b_lds,
                                 float (*p_lds)[16][16], float* c_lds,
                                 int nCol0, unsigned* bar, unsigned nb,
                                 unsigned gen) {
  const int tid  = threadIdx.x;
  const int lane = tid & 31;
  const int wave = tid >> 5;                  // 0..7
  const int g    = wave >> 1;                 // gate tile
  const int kh   = wave & 1;                  // K half
  const int m    = lane & 15;
  const int koff = (lane >> 4) << 1;
  {
    const float* hRow = hOld + (size_t)m * HID + kh * 512 + koff;
    const float* Brow = b_lds + (size_t)((g << 4) + m) * 1032;  // padded row
    v8f acc = {};
    for (int k = 0; k < 512; k += 4) {
      const int kk = k + kh * 512;
      const int kp = kk + koff + ((kk >> 8) << 1);
      v2f a = *(const v2f*)(hRow + k);
      v2f b = *(const v2f*)(Brow + kp);
      acc = wmma4(a, b, acc);
    }
    const int rHalf = (lane >> 4) << 3;
#pragma unroll
    for (int v = 0; v < 8; ++v) p_lds[wave][v + rHalf][m] = acc[v];
  }
  __syncthreads();
  {                                           // one element per thread
    const int mm = tid >> 4, n = tid & 15;
    const int col = nCol0 + n;
    const float* xp = Xp_t + (size_t)mm * GATES + col;
    const float iv = sigmoidf_(p_lds[0][mm][n] + p_lds[1][mm][n] + xp[0]);
    const float fv = sigmoidf_(p_lds[2][mm][n] + p_lds[3][mm][n] + xp[HID]);
    const float gv = tanhf    (p_lds[4][mm][n] + p_lds[5][mm][n] + xp[2 * HID]);
    const float ov = sigmoidf_(p_lds[6][mm][n] + p_lds[7][mm][n] + xp[3 * HID]);
    const float c  = fv * c_lds[tid] + iv * gv;
    c_lds[tid] = c;
    const float h = ov * tanhf(c);
    hNew[(size_t)mm * HID + col] = h;
    if (hsOut) hsOut[(size_t)mm * HID + col] = h;
  }
  grid_barrier(bar, nb, gen);
}

__global__ void __launch_bounds__(256, 1)
lstm_recurrence(const float* __restrict__ XPenc, const float* __restrict__ XPdec,
                const float* __restrict__ encWhh, const float* __restrict__ decWhh,
                float* __restrict__ hbuf, float* __restrict__ hs,
                unsigned* __restrict__ bar, unsigned nb) {
  extern __shared__ float b_lds[];            // 64 rows x 1032 floats (258 KB)
  __shared__ float p_lds[8][16][16];          // per-wave partial gate tiles
  __shared__ float c_lds[256];
  const int tid   = threadIdx.x;
  const int nCol0 = blockIdx.x * 16;          // this WGP owns h cols [nCol0,+16)
  unsigned gen = 0;

  hbuf[(size_t)(tid >> 4) * HID + nCol0 + (tid & 15)] = 0.0f;   // h0 = 0
  c_lds[tid] = 0.0f;                                            // c0 = 0
  // 3D TDM: rows {g*1024 + nCol0 + j}, j<16, g<4 -> 64 LDS rows (padded 1032)
  if (tid < 32) {
    tdm_load(lds_off_of(b_lds), encWhh + (size_t)nCol0 * HID, HID, 16u,
             (unsigned)GATES, 4u, (unsigned long long)HID * HID);
    __builtin_amdgcn_s_wait_tensorcnt(0);
  }
  __syncthreads();
  grid_barrier(bar, nb, ++gen);

  int p = 0;
  for (int t = 0; t < S_ENC; ++t) {
    lstm_step(XPenc + (size_t)t * NB * GATES, hbuf + (size_t)p * NB * HID,
              hbuf + (size_t)(p ^ 1) * NB * HID, nullptr,
              b_lds, p_lds, c_lds, nCol0, bar, nb, ++gen);
    p ^= 1;
  }
  __syncthreads();
  if (tid < 32) {                             // swap resident weights -> dec
    tdm_load(lds_off_of(b_lds), decWhh + (size_t)nCol0 * HID, HID, 16u,
             (unsigned)GATES, 4u, (unsigned long long)HID * HID);
    __builtin_amdgcn_s_wait_tensorcnt(0);
  }
  __syncthreads();
  for (int t = 0; t < T_DEC; ++t) {
    lstm_step(XPdec + (size_t)t * NB * GATES, hbuf + (size_t)p * NB * HID,
              hbuf + (size_t)(p ^ 1) * NB * HID, hs + (size_t)t * NB * HID,
              b_lds, p_lds, c_lds, nCol0, bar, nb, ++gen);
    p ^= 1;
  }
}

// ---------------------------------------------------------------- launch
extern "C" void kernel_launch(void* const* d_in, const int* in_sizes, int n_in,
                              void* d_out, int out_size, void* d_ws, size_t ws_size,
                              hipStream_t stream) {
  (void)in_sizes; (void)n_in; (void)out_size; (void)ws_size;
  const int*   enc_inputs = (const int*)d_in[0];
  const int*   dec_inputs = (const int*)d_in[1];
  const float* enc_emb = (const float*)d_in[2];
  const float* enc_Wih = (const float*)d_in[3];
  const float* enc_Whh = (const float*)d_in[4];
  const float* enc_bih = (const float*)d_in[5];
  const float* enc_bhh = (const float*)d_in[6];
  const float* dec_emb = (const float*)d_in[7];
  const float* dec_Wih = (const float*)d_in[8];
  const float* dec_Whh = (const float*)d_in[9];
  const float* dec_bih = (const float*)d_in[10];
  const float* dec_bhh = (const float*)d_in[11];
  const float* lin_W   = (const float*)d_in[12];
  const float* lin_b   = (const float*)d_in[13];
  float* out = (float*)d_out;

  char* ws = (char*)d_ws;
  unsigned* bar = (unsigned*)ws;                         ws += 256;
  float* Xenc  = (float*)ws; ws += sizeof(float) * (size_t)S_ENC * NB * EMB;
  float* Xdec  = (float*)ws; ws += sizeof(float) * (size_t)T_DEC * NB * EMB;
  float* XPenc = (float*)ws; ws += sizeof(float) * (size_t)S_ENC * NB * GATES;
  float* XPdec = (float*)ws; ws += sizeof(float) * (size_t)T_DEC * NB * GATES;
  float* hbuf  = (float*)ws; ws += sizeof(float) * (size_t)2 * NB * HID;
  float* hsbuf = (float*)ws; ws += sizeof(float) * (size_t)T_DEC * NB * HID;

  (void)hipMemsetAsync(bar, 0, sizeof(unsigned), stream); // reset grid barrier

  int tot4 = NB * S_ENC * (EMB / 4);
  embed_kernel<<<(tot4 + 255) / 256, 256, 0, stream>>>(enc_inputs, enc_emb, Xenc, S_ENC, tot4);
  tot4 = NB * T_DEC * (EMB / 4);
  embed_kernel<<<(tot4 + 255) / 256, 256, 0, stream>>>(dec_inputs, dec_emb, Xdec, T_DEC, tot4);

  // input projections: K=512 -> padded row 516 floats, B tile 132 KB
  {
    size_t ldsB = (size_t)64 * (EMB + ((EMB >> 8) << 1)) * sizeof(float) + 16;
    gemm_bT_kernel<<<dim3(GATES / 64, 4), 256, ldsB, stream>>>(
        Xenc, enc_Wih, enc_bih, enc_bhh, XPenc, S_ENC * NB, EMB, GATES, 0);
    gemm_bT_kernel<<<dim3(GATES / 64, 2), 256, ldsB, stream>>>(
        Xdec, dec_Wih, dec_bih, dec_bhh, XPdec, T_DEC * NB, EMB, GATES, 0);
  }

  lstm_recurrence<<<64, 256, (size_t)64 * 1032 * sizeof(float) + 16, stream>>>(
      XPenc, XPdec, enc_Whh, dec_Whh, hbuf, hsbuf, bar, 64u);

  // vocab projection: K=1024 -> padded row 1032 floats, B tile 258 KB,
  // one block per 64 vocab rows => lin_W read from HBM exactly once.
  {
    size_t ldsB = (size_t)64 * (HID + ((HID >> 8) << 1)) * sizeof(float) + 16;
    gemm_bT_kernel<<<dim3(VOCAB / 64, 1), 256, ldsB, stream>>>(
        hsbuf, lin_W, lin_b, nullptr, out, T_DEC * NB, HID, VOCAB, 1);
  }
}